// DAFXLayer_70918499991701
// MI455X (gfx1250) — compile-verified
//
#include <hip/hip_runtime.h>
#include <stdint.h>

typedef __attribute__((ext_vector_type(4))) float v4f;

#define THREADS 256
#define DEPTH   4                    // LDS slots (power of two)
#define INFLT   3                    // async stages in flight (< DEPTH: WAR-safe slot reuse)
#define ITERS   32                   // stages per block
#define STAGE_BYTES (THREADS * 16)   // 4 KiB per stage
#define STAGE_V4    (THREADS)        // v4f elements per stage (256)

// One 4 KiB stage: each lane async-copies its own 16 B to its own LDS slot.
// Tracked by ASYNCcnt; NT hint since the stream is touched exactly once.
__device__ __forceinline__ void async_stage(uint32_t lds_off, const v4f* gptr) {
  asm volatile("global_load_async_to_lds_b128 %0, %1, off th:TH_LOAD_NT"
               :: "v"(lds_off), "v"(gptr)
               : "memory");
}

__global__ __launch_bounds__(THREADS)
void dafx_gain_async(const float* __restrict__ x,
                     const float* __restrict__ params,
                     float* __restrict__ out, long T)
{
  __shared__ v4f buf[DEPTH][STAGE_V4];

  const int   tid = threadIdx.x;
  const int   b   = blockIdx.y;
  const float g   = params[b];                     // uniform -> scalar load

  const long  blockOff = (long)b * T + (long)blockIdx.x * (long)(ITERS * STAGE_V4 * 4);
  const v4f*  gbase = (const v4f*)(x + blockOff);
  v4f*        obase = (v4f*)(out + blockOff);

  // 32-bit LDS byte offset of this lane's slot (ptrtoint of an AS(3) pointer).
  typedef __attribute__((address_space(3))) v4f lds_v4f;
  const uint32_t lds0 = (uint32_t)(uintptr_t)(lds_v4f*)(&buf[0][0]) + (uint32_t)tid * 16u;

  // Prologue: put INFLT stages in flight.
#pragma unroll
  for (int s = 0; s < INFLT; ++s)
    async_stage(lds0 + (uint32_t)(s & (DEPTH - 1)) * STAGE_BYTES,
                gbase + (long)s * STAGE_V4 + tid);

  // Steady state: wait for stage i (in-order completion => cnt <= INFLT-1),
  // consume it, then prefetch stage i+INFLT into a slot last read at stage i-1.
  for (int i = 0; i < ITERS - INFLT; ++i) {
    asm volatile("s_wait_asynccnt 2" ::: "memory");
    v4f v = buf[i & (DEPTH - 1)][tid];
    v4f r = v * g;
    __builtin_nontemporal_store(r, obase + (long)i * STAGE_V4 + tid);

    const int pf = i + INFLT;
    async_stage(lds0 + (uint32_t)(pf & (DEPTH - 1)) * STAGE_BYTES,
                gbase + (long)pf * STAGE_V4 + tid);
  }

  // Epilogue: drain last INFLT stages with decreasing waits.
  {
    int s = ITERS - 3;
    asm volatile("s_wait_asynccnt 2" ::: "memory");
    v4f r0 = buf[s & (DEPTH - 1)][tid] * g;
    __builtin_nontemporal_store(r0, obase + (long)s * STAGE_V4 + tid);

    s = ITERS - 2;
    asm volatile("s_wait_asynccnt 1" ::: "memory");
    v4f r1 = buf[s & (DEPTH - 1)][tid] * g;
    __builtin_nontemporal_store(r1, obase + (long)s * STAGE_V4 + tid);

    s = ITERS - 1;
    asm volatile("s_wait_asynccnt 0" ::: "memory");
    v4f r2 = buf[s & (DEPTH - 1)][tid] * g;
    __builtin_nontemporal_store(r2, obase + (long)s * STAGE_V4 + tid);
  }
}

// Generic fallback for shapes that don't match the tiling (grid-stride).
__global__ void dafx_gain_simple(const float* __restrict__ x,
                                 const float* __restrict__ params,
                                 float* __restrict__ out, long T, long n)
{
  long i      = (long)blockIdx.x * blockDim.x + threadIdx.x;
  long stride = (long)gridDim.x * blockDim.x;
  for (; i < n; i += stride)
    out[i] = x[i] * params[i / T];
}

extern "C" void kernel_launch(void* const* d_in, const int* in_sizes, int n_in,
                              void* d_out, int out_size, void* d_ws, size_t ws_size,
                              hipStream_t stream) {
  const float* x = (const float*)d_in[0];
  const float* p = (const float*)d_in[1];
  float*     out = (float*)d_out;

  const long n = (long)in_sizes[0];
  const long B = (n_in > 1 && in_sizes[1] > 0) ? (long)in_sizes[1] : 1;
  const long T = n / B;

  const long floatsPerBlock = (long)ITERS * STAGE_V4 * 4;   // 32768 floats
  if (T > 0 && (T % floatsPerBlock) == 0 &&
      (T / floatsPerBlock) <= 65535 && B <= 65535 && (n % B) == 0) {
    dim3 grid((unsigned)(T / floatsPerBlock), (unsigned)B);
    dafx_gain_async<<<grid, dim3(THREADS), 0, stream>>>(x, p, out, T);
  } else {
    long blocks = (n + THREADS - 1) / THREADS;
    if (blocks > 8192) blocks = 8192;
    if (blocks < 1) blocks = 1;
    dafx_gain_simple<<<dim3((unsigned)blocks), dim3(THREADS), 0, stream>>>(x, p, out, T, n);
  }
}